// DensityAwareChamferReward_14757507629949
// MI455X (gfx1250) — compile-verified
//
#include <hip/hip_runtime.h>

// CDNA5 / gfx1250, wave32. One workgroup (256 threads = 8 waves) per (b,v).
// Cross term of the pairwise sq-distance matrix computed with
// V_WMMA_F32_16X16X4_F32 (K=4 matches the 4 visibility features exactly).
// Each wave owns 4 fixed row tiles (A fragments + row norms preloaded in
// registers) and sweeps the 32 column tiles: per column tile, 3 LDS loads
// feed 4 WMMAs, column-min keys accumulate in-register across the 4 row
// tiles (1 ds_min_u64 per lane per column tile), row-min keys live entirely
// in registers until a final 16-lane butterfly reduce.

typedef __attribute__((ext_vector_type(2))) float v2f;
typedef __attribute__((ext_vector_type(8))) float v8f;

#define NPTS 512
#define THRESH_SQ 6.0f

__device__ __forceinline__ unsigned ordered_bits(float f) {
    unsigned u = __float_as_uint(f);
    return u ^ ((u >> 31) ? 0xFFFFFFFFu : 0x80000000u);
}
__device__ __forceinline__ float unordered_bits(unsigned v) {
    unsigned u = (v & 0x80000000u) ? (v ^ 0x80000000u) : ~v;
    return __uint_as_float(u);
}

// Computes r = -(sum(w*dist) + unmatched) / max(groups + unmatched, 1)
// key[q]: (ordered_dist<<32)|src_idx for query q. Valid return only on tid==0.
__device__ float direction_r(const unsigned long long* key,
                             const float* qxy, const float* srcxy,
                             int* cnt, float* red, int* sg, int* su, int tid) {
    if (tid == 0) { *sg = 0; *su = 0; }
    // phase A: scatter-count matched queries onto their nearest source
    for (int q = tid; q < NPTS; q += 256) {
        unsigned long long k = key[q];
        float md = unordered_bits((unsigned)(k >> 32));
        if (!(md > THRESH_SQ)) {
            int idx = (int)(k & 0xFFFFFFFFu);
            atomicAdd(&cnt[idx], 1);
        }
    }
    __syncthreads();
    // phase B: weighted distance sum, group count, unmatched flag
    float local = 0.0f;
    int lat = 0, grp = 0;
    for (int q = tid; q < NPTS; q += 256) {
        unsigned long long k = key[q];
        float md = unordered_bits((unsigned)(k >> 32));
        int idx = (int)(k & 0xFFFFFFFFu);
        if (md > THRESH_SQ) {
            lat = 1;
        } else {
            float dx = qxy[2 * q]     - srcxy[2 * idx];
            float dy = qxy[2 * q + 1] - srcxy[2 * idx + 1];
            float dist = sqrtf(dx * dx + dy * dy);
            float w = 1.0f / ((float)cnt[idx] + 1e-6f);
            local += w * dist;
        }
        if (cnt[q] > 0) grp++;   // q doubles as source-slot index j
    }
    atomicAdd(sg, grp);          // integer: deterministic
    if (lat) atomicOr(su, 1);
    red[tid] = local;
    __syncthreads();
    for (int s = 128; s > 0; s >>= 1) {       // fixed-pattern float reduce
        if (tid < s) red[tid] += red[tid + s];
        __syncthreads();
    }
    float r = 0.0f;
    if (tid == 0) {
        float unm = (*su) ? 1.0f : 0.0f;
        float groups = fmaxf((float)(*sg) + unm, 1.0f);
        r = -(red[0] + unm) / groups;
    }
    __syncthreads();             // protect red/sg/su for next call
    return r;
}

__global__ __launch_bounds__(256)
void chamfer_reward_kernel(const float* __restrict__ ach,
                           const float* __restrict__ des,
                           float* __restrict__ out) {
    __shared__ float sv[NPTS * 4];    // s_vis
    __shared__ float gv[NPTS * 4];    // g_vis
    __shared__ float sxy[NPTS * 2];
    __shared__ float gxy[NPTS * 2];
    __shared__ float ssq[NPTS];
    __shared__ float gsq[NPTS];
    __shared__ unsigned long long rowkey[NPTS];  // md2/mi2: min over m per n
    __shared__ unsigned long long colkey[NPTS];  // md1/mi1: min over n per m
    __shared__ int cntA[NPTS];
    __shared__ int cntB[NPTS];
    __shared__ float red[256];
    __shared__ int sg, su;

    const int bv   = blockIdx.x;       // (b*2 + v)
    const int b    = bv >> 1;
    const int tid  = threadIdx.x;
    const int lane = tid & 31;
    const int wave = tid >> 5;
    const bool hi  = lane >= 16;
    const int l16  = lane & 15;

    const float* A = ach + (size_t)bv * (NPTS * 10);
    const float* G = des + (size_t)bv * (NPTS * 10);

    // ---- stage inputs into LDS, compute squared norms of vis features ----
    for (int r = tid; r < NPTS; r += 256) {
        const float* ar = A + r * 10;
        float a5 = ar[5], a6 = ar[6], a7 = ar[7], a8 = ar[8];
        sxy[2 * r] = ar[0]; sxy[2 * r + 1] = ar[1];
        sv[4 * r] = a5; sv[4 * r + 1] = a6; sv[4 * r + 2] = a7; sv[4 * r + 3] = a8;
        ssq[r] = a5 * a5 + a6 * a6 + a7 * a7 + a8 * a8;

        const float* gr = G + r * 10;
        float g5 = gr[5], g6 = gr[6], g7 = gr[7], g8 = gr[8];
        gxy[2 * r] = gr[0]; gxy[2 * r + 1] = gr[1];
        gv[4 * r] = g5; gv[4 * r + 1] = g6; gv[4 * r + 2] = g7; gv[4 * r + 3] = g8;
        gsq[r] = g5 * g5 + g6 * g6 + g7 * g7 + g8 * g8;

        colkey[r] = ~0ULL;
        cntA[r] = 0; cntB[r] = 0;
    }
    __syncthreads();

    // ---- preload this wave's 4 row tiles: A fragments + row norms ----
    // A fragment 16x4 f32: lanes 0-15 hold K=0,1; lanes 16-31 hold K=2,3.
    v2f afr[4];
    float rs[4][8];
    unsigned long long rk[4][8];
    #pragma unroll
    for (int t = 0; t < 4; ++t) {
        const int n0 = (wave + 8 * t) * 16;
        const int row = n0 + l16;
        const int k0 = hi ? 2 : 0;
        afr[t].x = sv[4 * row + k0];
        afr[t].y = sv[4 * row + k0 + 1];
        #pragma unroll
        for (int vg = 0; vg < 8; ++vg) {
            rs[t][vg] = ssq[n0 + vg + (hi ? 8 : 0)];
            rk[t][vg] = ~0ULL;
        }
    }

    // ---- sweep column tiles: 3 LDS loads feed 4 WMMAs each ----
    for (int tc = 0; tc < NPTS / 16; ++tc) {
        const int m0 = tc * 16;
        const int mcol = m0 + l16;
        // B fragment 4x16 f32: VGPR j = K=j (lanes 0-15) / K=j+2 (lanes 16-31)
        v2f bfr;
        bfr.x = gv[4 * mcol + (hi ? 2 : 0)];
        bfr.y = gv[4 * mcol + (hi ? 3 : 1)];
        const float gq = gsq[mcol];

        unsigned long long ck = ~0ULL;   // col-min across all 4 row tiles
        #pragma unroll
        for (int t = 0; t < 4; ++t) {
            const int n0 = (wave + 8 * t) * 16;
            v8f c = {};
            c = __builtin_amdgcn_wmma_f32_16x16x4_f32(
                    false, afr[t], false, bfr, (short)0, c, false, false);
            #pragma unroll
            for (int vg = 0; vg < 8; ++vg) {
                float p = rs[t][vg] + gq - 2.0f * c[vg];   // ||s-g||^2
                unsigned long long ub = (unsigned long long)ordered_bits(p) << 32;
                const int row = n0 + vg + (hi ? 8 : 0);
                unsigned long long kr = ub | (unsigned)mcol;  // min over m
                if (kr < rk[t][vg]) rk[t][vg] = kr;
                unsigned long long kc = ub | (unsigned)row;   // min over n
                if (kc < ck) ck = kc;
            }
        }
        atomicMin(&colkey[mcol], ck);   // ds_min_u64, shared across waves
    }

    // rows owned by this wave: butterfly min across the 16-lane N group
    #pragma unroll
    for (int t = 0; t < 4; ++t) {
        const int n0 = (wave + 8 * t) * 16;
        #pragma unroll
        for (int vg = 0; vg < 8; ++vg) {
            unsigned long long k = rk[t][vg];
            #pragma unroll
            for (int off = 1; off < 16; off <<= 1) {
                unsigned long long o = __shfl_xor(k, off, 16);
                if (o < k) k = o;
            }
            if (l16 == 0) rowkey[n0 + vg + (hi ? 8 : 0)] = k;
        }
    }
    __syncthreads();

    // ---- _direction for both matchings (masks are constant-False) ----
    float r_g2s = direction_r(colkey, gxy, sxy, cntA, red, &sg, &su, tid);
    float r_s2g = direction_r(rowkey, sxy, gxy, cntB, red, &sg, &su, tid);

    // rv = (r_g2s + r_s2g)/2 ; reward[b] = mean over v (2 slices) -> /4 total
    if (tid == 0) atomicAdd(&out[b], 0.25f * (r_g2s + r_s2g));
}

__global__ void zero_kernel(float* __restrict__ out, int n) {
    int i = blockIdx.x * blockDim.x + threadIdx.x;
    if (i < n) out[i] = 0.0f;
}

extern "C" void kernel_launch(void* const* d_in, const int* in_sizes, int n_in,
                              void* d_out, int out_size, void* d_ws, size_t ws_size,
                              hipStream_t stream) {
    (void)in_sizes; (void)n_in; (void)d_ws; (void)ws_size;
    const float* achieved = (const float*)d_in[0];
    const float* desired  = (const float*)d_in[1];
    float* out = (float*)d_out;

    zero_kernel<<<(out_size + 255) / 256, 256, 0, stream>>>(out, out_size);
    chamfer_reward_kernel<<<dim3(512), dim3(256), 0, stream>>>(achieved, desired, out);
}